// SSIMLoss_75642964017454
// MI455X (gfx1250) — compile-verified
//
#include <hip/hip_runtime.h>
#include <hip/hip_bf16.h>

typedef __attribute__((ext_vector_type(2))) float v2f;
typedef __attribute__((ext_vector_type(8))) float v8f;

#define B_   16
#define C_   3
#define H_   512
#define W_   1024
#define CW   256          // crop width
#define TH   16           // output tile rows
#define TW   64           // output tile cols
#define KR   26           // raw rows = TH + 10
#define RW   74           // raw cols = TW + 10
#define RSTRIDE 76
#define KP   28           // K padded to multiple of 4
#define TSTRIDE 72        // tmp col stride (2*72 % 64 == 16 -> conflict-free +-2 row fetch)
#define NTOT (16.0 * 3.0 * 512.0 * 256.0)

// 11-tap Gaussian, sigma = 1.5, normalized (matches numpy reference)
__device__ __constant__ float G11[11] = {
    0.00102838f, 0.00759876f, 0.03600077f, 0.10936069f, 0.21300564f,
    0.26601172f,
    0.21300564f, 0.10936069f, 0.03600077f, 0.00759876f, 0.00102838f};

__global__ void ssim_zero(double* acc) { *acc = 0.0; }

__global__ __launch_bounds__(128)
void ssim_main(const float* __restrict__ img1, const float* __restrict__ img2,
               const int* __restrict__ pos, double* __restrict__ acc)
{
    __shared__ float raw1[KR][RSTRIDE];
    __shared__ float raw2[KR][RSTRIDE];
    __shared__ float tmp[5][KP][TSTRIDE];

    const int bid    = blockIdx.x;
    const int cchunk = bid & 3;          // 4 col tiles of 64
    const int rtile  = (bid >> 2) & 31;  // 32 row tiles of 16
    const int ch     = (bid >> 7) % 3;
    const int b      = bid / 384;
    const int c0     = cchunk * TW;
    const int r0     = rtile * TH;

    int s = pos[b];
    s = s < 0 ? 0 : (s > (W_ - CW) ? (W_ - CW) : s);

    const float* base1 = img1 + (size_t)(b * C_ + ch) * H_ * W_;
    const float* base2 = img2 + (size_t)(b * C_ + ch) * H_ * W_;

    // ---- stage halo tiles (zero-padded relative to the CROPPED image) ----
    for (int idx = threadIdx.x; idx < KR * RW; idx += 128) {
        const int k  = idx / RW;
        const int j  = idx % RW;
        const int row = r0 - 5 + k;
        const int cc  = c0 - 5 + j;      // column in cropped coords
        float a1 = 0.0f, a2 = 0.0f;
        if (row >= 0 && row < H_ && cc >= 0 && cc < CW) {
            const size_t off = (size_t)row * W_ + (size_t)(s + cc);
            a1 = base1[off];
            a2 = base2[off];
        }
        raw1[k][j] = a1;
        raw2[k][j] = a2;
    }
    __syncthreads();

    // ---- horizontal 11-tap conv of 5 product maps; rows 26..27 zeroed ----
    for (int idx = threadIdx.x; idx < KP * TW; idx += 128) {
        const int k = idx / TW;
        const int c = idx % TW;
        float s0 = 0.f, s1 = 0.f, s2 = 0.f, s3 = 0.f, s4 = 0.f;
        if (k < KR) {
#pragma unroll
            for (int j = 0; j < 11; ++j) {
                const float w  = G11[j];
                const float a1 = raw1[k][c + j];
                const float a2 = raw2[k][c + j];
                s0 = fmaf(w, a1, s0);
                s1 = fmaf(w, a2, s1);
                s2 = fmaf(w * a1, a1, s2);
                s3 = fmaf(w * a2, a2, s3);
                s4 = fmaf(w * a1, a2, s4);
            }
        }
        tmp[0][k][c] = s0;
        tmp[1][k][c] = s1;
        tmp[2][k][c] = s2;
        tmp[3][k][c] = s3;
        tmp[4][k][c] = s4;
    }
    __syncthreads();

    // ---- vertical conv via chained V_WMMA_F32_16X16X4_F32 ----
    const int wave = threadIdx.x >> 5;   // 4 waves -> 4 chunks of 16 cols
    const int lane = threadIdx.x & 31;
    const int m    = lane & 15;          // A-matrix row / D col index helper
    const int koff = (lane < 16) ? 0 : 2;
    const int n    = wave * 16 + m;      // tmp column this lane feeds (B col)

    // A[m][k] = g[k - m] for k in [m, m+10], banded 16x28 (zero K-pad)
    float ax[7], ay[7];
#pragma unroll
    for (int kk = 0; kk < 7; ++kk) {
        const int k0 = 4 * kk + koff;
        const int d0 = k0 - m;
        const int d1 = k0 + 1 - m;
        ax[kk] = (d0 >= 0 && d0 <= 10) ? G11[d0] : 0.0f;
        ay[kk] = (d1 >= 0 && d1 <= 10) ? G11[d1] : 0.0f;
    }

    v8f conv[5];
#pragma unroll
    for (int map = 0; map < 5; ++map) {
        v8f d = {0.f, 0.f, 0.f, 0.f, 0.f, 0.f, 0.f, 0.f};
#pragma unroll
        for (int kk = 0; kk < 7; ++kk) {
            v2f a, bb;
            a.x  = ax[kk];
            a.y  = ay[kk];
            bb.x = tmp[map][4 * kk + koff][n];
            bb.y = tmp[map][4 * kk + 1 + koff][n];
            d = __builtin_amdgcn_wmma_f32_16x16x4_f32(
                    false, a, false, bb, (short)0, d, false, false);
        }
        conv[map] = d;
    }

    // ---- SSIM map + reduction (lane holds 8 elements of its 16x16 tile) ----
    const float C1 = 1e-4f, C2 = 9e-4f, EPS = 1e-5f;
    float partial = 0.0f;
#pragma unroll
    for (int i = 0; i < 8; ++i) {
        const float mu1  = conv[0][i];
        const float mu2  = conv[1][i];
        const float e11  = conv[2][i];
        const float e22  = conv[3][i];
        const float e12  = conv[4][i];
        const float mu1s = mu1 * mu1;
        const float mu2s = mu2 * mu2;
        const float mu12 = mu1 * mu2;
        const float sg1  = e11 - mu1s;
        const float sg2  = e22 - mu2s;
        const float sg12 = e12 - mu12;
        const float num = (2.0f * mu12 + C1) * (2.0f * sg12 + C2) + EPS;
        const float den = (mu1s + mu2s + C1) * (sg1 + sg2 + C2) + EPS;
        partial += num / den;
    }
#pragma unroll
    for (int off = 16; off > 0; off >>= 1)
        partial += __shfl_xor(partial, off, 32);
    if (lane == 0)
        atomicAdd(acc, (double)partial);
}

__global__ void ssim_final(const double* acc, float* out)
{
    *out = 1.0f - (float)(*acc / NTOT);
}

extern "C" void kernel_launch(void* const* d_in, const int* in_sizes, int n_in,
                              void* d_out, int out_size, void* d_ws, size_t ws_size,
                              hipStream_t stream)
{
    const float* img1 = (const float*)d_in[0];
    const float* img2 = (const float*)d_in[1];
    const int*   pos  = (const int*)d_in[2];   // jax default x64-off -> int32
    double* acc = (double*)d_ws;
    float*  out = (float*)d_out;

    ssim_zero<<<1, 1, 0, stream>>>(acc);
    // B*C*(H/16)*(256/64) = 16*3*32*4 = 6144 tiles
    ssim_main<<<6144, 128, 0, stream>>>(img1, img2, pos, acc);
    ssim_final<<<1, 1, 0, stream>>>(acc, out);
}